// IcosahedronConv2d_15960098472015
// MI455X (gfx1250) — compile-verified
//
#include <hip/hip_runtime.h>
#include <hip/hip_bf16.h>

typedef __attribute__((ext_vector_type(2))) float v2f;
typedef __attribute__((ext_vector_type(8))) float v8f;

#define NV    10242            // icosphere vertices
#define NB    8                // batch
#define CIN   128
#define COUT  128
#define MTOT  (NB * NV)        // 81936 = 144 * 569
#define ROWS  144              // M rows per block (9 tiles of 16)
#define TILES 9                // M tiles per wave
#define AS    132              // LDS A row stride (words): conflict-free frag reads
#define BS    136              // LDS B row stride (words): lane halves hit disjoint banks
#define WELEM (9 * CIN * COUT) // 147456 packed weight elements

// Pack w[o][c][kk] (kk innermost) -> wp[kk][c][o] for coalesced float4 staging.
__global__ __launch_bounds__(256) void pack_w_kernel(const float* __restrict__ w,
                                                     float* __restrict__ wp) {
    int e = blockIdx.x * 256 + threadIdx.x;
    if (e < WELEM) {
        int kk  = e / (CIN * COUT);
        int rem = e - kk * (CIN * COUT);
        int c   = rem >> 7;
        int o   = rem & 127;
        wp[e] = w[(o * CIN + c) * 9 + kk];
    }
}

// out[m][o] = sum_{kk,c} x[b][idx[v][kk]][c]*mask[v][kk] * w[o][c][kk] + bias[o]
// m = b*NV + v.  Block: 144 rows x 128 cols; 8 waves, each wave owns a 16-col
// slice across 9 M-tiles, accumulated with V_WMMA_F32_16X16X4_F32.
__global__ __launch_bounds__(256) void icoconv_wmma_kernel(
    const float* __restrict__ x, const float* __restrict__ wsrc,
    const float* __restrict__ bias, const int* __restrict__ nidx,
    const float* __restrict__ nmask, float* __restrict__ out, int packed) {

    __shared__ float lA[ROWS * AS];   // 144 x 128 A panel (padded)     ~76.0 KB
    __shared__ float lB[CIN * BS];    // 128 x 128 B panel (padded)     ~69.6 KB

    const int tid  = threadIdx.x;
    const int wq   = tid >> 5;        // wave id (also staging row group)
    const int lane = tid & 31;
    const int half = lane >> 4;       // 0: K=k0,k0+1 ; 1: K=k0+2,k0+3
    const int l16  = lane & 15;
    const int col0 = wq * 16;         // this wave's N slice
    const int c4   = tid & 31;        // per-thread float4 column (constant)

    const int mbase = blockIdx.x * ROWS;      // < 81936, fits int
    const int b0 = mbase / NV;                // uniform -> scalar div, once
    const int v0 = mbase - b0 * NV;

    v8f acc[TILES];
    #pragma unroll
    for (int t = 0; t < TILES; ++t) acc[t] = (v8f){0.f,0.f,0.f,0.f,0.f,0.f,0.f,0.f};

    const float4* x4 = reinterpret_cast<const float4*>(x);

    for (int kk = 0; kk < 9; ++kk) {
        __syncthreads();  // previous iteration's LDS reads done before restage

        // ---- stage A: gather 144 masked neighbor rows of x, float4 coalesced.
        // slot = tid + i*256  ->  r = wq + 8*i, col4 = c4 (constant).
        #pragma unroll
        for (int i = 0; i < 18; ++i) {
            int r = wq + 8 * i;
            int v = v0 + r;
            int b = b0;
            if (v >= NV) { v -= NV; b += 1; }
            int   j  = nidx[v * 9 + kk];
            float ms = nmask[v * 9 + kk];
            float4 val = x4[((long)((b * NV + j) << 5)) + c4];
            float4 sv;
            sv.x = val.x * ms; sv.y = val.y * ms; sv.z = val.z * ms; sv.w = val.w * ms;
            *reinterpret_cast<float4*>(&lA[r * AS + c4 * 4]) = sv;
        }

        // ---- stage B: 128x128 weight slice for this kk into [c][o] layout
        if (packed) {
            const float4* wp4 = reinterpret_cast<const float4*>(wsrc) + (size_t)kk * 4096;
            #pragma unroll
            for (int i = 0; i < 16; ++i) {
                int c = wq + 8 * i;                 // (tid + i*256)>>5
                float4 val = wp4[c * 32 + c4];
                *reinterpret_cast<float4*>(&lB[c * BS + c4 * 4]) = val;
            }
        } else {
            #pragma unroll
            for (int i = 0; i < 64; ++i) {
                int c = (tid >> 7) + 2 * i;         // (tid + i*256)>>7
                int o = tid & 127;
                lB[c * BS + o] = wsrc[(o * CIN + c) * 9 + kk];
            }
        }

        __syncthreads();

        // ---- compute: 32 k-steps of 4; one B frag feeds 9 WMMAs
        #pragma unroll 2
        for (int k0 = 0; k0 < CIN; k0 += 4) {
            const int ck = k0 + 2 * half;

            v2f bf;                                  // B frag: rows ck, ck+1 @ col0+l16
            bf[0] = lB[ck * BS + col0 + l16];
            bf[1] = lB[(ck + 1) * BS + col0 + l16];

            #pragma unroll
            for (int t = 0; t < TILES; ++t) {
                v2f a;                               // A frag: row t*16+l16, cols ck, ck+1
                a[0] = lA[(t * 16 + l16) * AS + ck];
                a[1] = lA[(t * 16 + l16) * AS + ck + 1];
                acc[t] = __builtin_amdgcn_wmma_f32_16x16x4_f32(
                    false, a, false, bf, (short)0, acc[t], false, false);
            }
        }
    }

    // ---- epilogue: C layout is VGPR i -> M = i + 8*half, N = l16
    const float bv = bias[col0 + l16];
    float* outp = out + (long)(mbase + 8 * half) * COUT + col0 + l16;
    #pragma unroll
    for (int t = 0; t < TILES; ++t) {
        #pragma unroll
        for (int i = 0; i < 8; ++i) {
            outp[(long)(t * 16 + i) * COUT] = acc[t][i] + bv;
        }
    }
}

extern "C" void kernel_launch(void* const* d_in, const int* in_sizes, int n_in,
                              void* d_out, int out_size, void* d_ws, size_t ws_size,
                              hipStream_t stream) {
    const float* x     = (const float*)d_in[0];  // [8, 10242, 128] f32
    const float* w     = (const float*)d_in[1];  // [128, 128, 3, 3] f32
    const float* bias  = (const float*)d_in[2];  // [128] f32
    const int*   nidx  = (const int*)  d_in[3];  // [10242, 9] i32
    const float* nmask = (const float*)d_in[4];  // [10242, 9] f32
    float*       out   = (float*)d_out;          // [8, 10242, 128] f32

    const size_t need = (size_t)WELEM * sizeof(float);
    const float* wsrc = w;
    int packed = 0;
    if (ws_size >= need) {
        float* wp = (float*)d_ws;
        pack_w_kernel<<<(WELEM + 255) / 256, 256, 0, stream>>>(w, wp);
        wsrc = wp;
        packed = 1;
    }

    const int nblocks = MTOT / ROWS;  // 569, exact
    icoconv_wmma_kernel<<<nblocks, 256, 0, stream>>>(x, wsrc, bias, nidx, nmask, out, packed);
}